// AFNO2D_54305566491205
// MI455X (gfx1250) — compile-verified
//
#include <hip/hip_runtime.h>
#include <math.h>

// ---------------- problem constants (match reference) ----------------
#define HH   128
#define WW   128
#define WF   65      // rfft width
#define WFL  80      // logical padded width (5 x 16 WMMA tiles)
#define WFP  84      // LDS pitch for 80-wide tiles (bank-conflict free, 16B aligned)
#define XSP  132     // LDS pitch for 128-wide tiles
#define ZP   68      // LDS pitch for 64-wide mixer tiles
#define FTP  36      // LDS pitch for 32-wide fuse A tiles
#define CH   512
#define NBK  8
#define BSZ  64
#define M1   64
#define M2   33
#define NPTS (M1*M2) // 2112 masked modes
#define LAMBDA 0.01f

typedef float v2f __attribute__((ext_vector_type(2)));
typedef float v8f __attribute__((ext_vector_type(8)));

// ---------------- async global->LDS path (CDNA5), guarded -------------
// Round-2 probe: param 0 is int4* in address_space(1) (global src, non-const),
// param 1 is the LDS destination, trailing args are constant ints.
#if defined(__gfx1250__) && \
    __has_builtin(__builtin_amdgcn_global_load_async_to_lds_b128) && \
    __has_builtin(__builtin_amdgcn_s_wait_asynccnt)
#define HAVE_ASYNC 1
typedef int v4i __attribute__((ext_vector_type(4)));
typedef __attribute__((address_space(1))) v4i gv4i;
typedef __attribute__((address_space(3))) v4i lv4i;
#define ASYNC_CP_F4(dst, src) \
  __builtin_amdgcn_global_load_async_to_lds_b128((gv4i*)(src), (lv4i*)(dst), 0, 0)
#define ASYNC_FENCE() __builtin_amdgcn_s_wait_asynccnt(0)
#else
#define HAVE_ASYNC 0
#define ASYNC_CP_F4(dst, src) (*(float4*)(dst) = *(const float4*)(src))
#define ASYNC_FENCE()
#endif

// ---------------------------------------------------------------------
// Accumulate C += A(16xK) * B(Kx16) with V_WMMA_F32_16X16X4_F32.
// Fragment layouts per CDNA5 ISA 7.12.2:
//   A 16x4 f32 : VGPR0 = {K=k0 (lanes 0-15), K=k0+2 (lanes 16-31)}, VGPR1 = +1
//   B 4x16 f32 : mirrored; C/D 16x16: VGPR j = row M=j / M=j+8 per lane half
// A, B may live in LDS or global (flat addressing). EXEC must be full.
// ---------------------------------------------------------------------
__device__ __forceinline__ void wmma_acc_f32(v8f& acc,
    const float* __restrict__ A, int lda,
    const float* __restrict__ B, int ldb, int K)
{
  const int l  = threadIdx.x & 15;
  const int hf = (threadIdx.x >> 4) & 1;
  const float* Arow = A + l * lda + 2 * hf;
  const float* Bcol = B + l + (2 * hf) * ldb;
  for (int k0 = 0; k0 < K; k0 += 4) {
    v2f a, b;
    a.x = Arow[k0];
    a.y = Arow[k0 + 1];
    b.x = Bcol[k0 * ldb];
    b.y = Bcol[k0 * ldb + ldb];
    acc = __builtin_amdgcn_wmma_f32_16x16x4_f32(
        false, a, false, b, (short)0, acc, false, false);
  }
}

// ---------------------------------------------------------------------
// DFT twiddle tables: COS[k][n] = cos(2*pi*k*n/128), SIN likewise (symmetric).
// ---------------------------------------------------------------------
__global__ void k_init_dft(float* __restrict__ tab)
{
  int k = blockIdx.x, n = threadIdx.x;
  int kn = (k * n) & 127;
  float ang = 6.28318530717958647692f * (float)kn * (1.0f / 128.0f);
  tab[k * 128 + n]         = cosf(ang);
  tab[16384 + k * 128 + n] = sinf(ang);
}

// ---------------------------------------------------------------------
// rfft2 (ortho): one workgroup per (b,c) image.
// ---------------------------------------------------------------------
__global__ void k_rfft2(const float* __restrict__ x, const float* __restrict__ tab,
                        float* __restrict__ Zr, float* __restrict__ Zi)
{
  extern __shared__ float sm[];
  float* xs = sm;                    // 128 x XSP
  float* tr = sm + HH * XSP;         // 128 x WFP
  float* ti = tr + HH * WFP;
  const float* COS = tab;
  const float* SIN = tab + 16384;

  const int img = blockIdx.x;
  const float* xp = x + (size_t)img * (HH * WW);
  const int tid = threadIdx.x, wv = tid >> 5;
  const int l = tid & 15, h8 = ((tid >> 4) & 1) * 8;

  for (int i = tid; i < (HH * WW) / 4; i += 256) {
    int row = i >> 5, q = (i & 31) * 4;
    ASYNC_CP_F4(xs + row * XSP + q, xp + row * WW + q);
  }
  ASYNC_FENCE();
  __syncthreads();

  // stage 1 (rows): 8(h) x 5(k) tiles, real/imag -> 80 wave jobs
  for (int t = wv; t < 80; t += 8) {
    int ri = t & 1, tt = t >> 1;
    int hm = (tt / 5) << 4, kn = (tt % 5) << 4;
    v8f acc = {};
    wmma_acc_f32(acc, xs + hm * XSP, XSP, (ri ? SIN : COS) + kn, 128, 128);
    float* dst = ri ? ti : tr;
    float sg = ri ? -1.0f : 1.0f;   // imag part of e^{-i theta}
    for (int j = 0; j < 8; ++j)
      dst[(hm + j + h8) * WFP + kn + l] = sg * acc[j];
  }
  __syncthreads();

  // stage 2 (columns): complex DFT, 8(k1) x 5(k2) tiles
  float* zrp = Zr + (size_t)img * (HH * WF);
  float* zip = Zi + (size_t)img * (HH * WF);
  for (int t = wv; t < 40; t += 8) {
    int km = (t / 5) << 4, kn = (t % 5) << 4;
    v8f aR = {}, bR = {}, aI = {}, bI = {};
    wmma_acc_f32(aR, COS + km * 128, 128, tr + kn, WFP, 128);
    wmma_acc_f32(bR, SIN + km * 128, 128, ti + kn, WFP, 128);
    wmma_acc_f32(aI, COS + km * 128, 128, ti + kn, WFP, 128);
    wmma_acc_f32(bI, SIN + km * 128, 128, tr + kn, WFP, 128);
    if (kn + l < WF) {
      for (int j = 0; j < 8; ++j) {
        int row = km + j + h8;
        zrp[row * WF + kn + l] = (aR[j] + bR[j]) * (1.0f / 128.0f);
        zip[row * WF + kn + l] = (aI[j] - bI[j]) * (1.0f / 128.0f);
      }
    }
  }
}

// ---------------------------------------------------------------------
// Block-diagonal complex MLP on masked modes (in-place on Zr/Zi).
// ---------------------------------------------------------------------
__global__ void k_mixer(float* __restrict__ Zr, float* __restrict__ Zi,
    const float* __restrict__ W1r, const float* __restrict__ W1i,
    const float* __restrict__ B1r, const float* __restrict__ B1i,
    const float* __restrict__ W2r, const float* __restrict__ W2i,
    const float* __restrict__ B2r, const float* __restrict__ B2i)
{
  extern __shared__ float sm[];
  float* w1r = sm;                 // 64*64 each (B-side: no pad needed)
  float* w1i = w1r + 4096;
  float* w2r = w1i + 4096;
  float* w2i = w2r + 4096;
  float* zr  = w2i + 4096;         // 32 x ZP (A-side: padded)
  float* zi  = zr + 32 * ZP;
  float* hr  = zi + 32 * ZP;
  float* hi  = hr + 32 * ZP;

  const int b = blockIdx.z, n = blockIdx.y, p0 = blockIdx.x * 32;
  const int tid = threadIdx.x, wv = tid >> 5;
  const int l = tid & 15, h8 = ((tid >> 4) & 1) * 8;

  for (int i = tid; i < 1024; i += 256) {
    ASYNC_CP_F4(w1r + 4 * i, W1r + n * 4096 + 4 * i);
    ASYNC_CP_F4(w1i + 4 * i, W1i + n * 4096 + 4 * i);
    ASYNC_CP_F4(w2r + 4 * i, W2r + n * 4096 + 4 * i);
    ASYNC_CP_F4(w2i + 4 * i, W2i + n * 4096 + 4 * i);
  }
  const size_t chbase = ((size_t)b * CH + n * BSZ) * (HH * WF);
  for (int i = tid; i < 2048; i += 256) {
    int r = i >> 6, ch = i & 63;
    int p = p0 + r, k1 = p / M2, k2 = p % M2;
    size_t off = chbase + (size_t)ch * (HH * WF) + k1 * WF + k2;
    zr[r * ZP + ch] = Zr[off];
    zi[r * ZP + ch] = Zi[off];
  }
  ASYNC_FENCE();
  __syncthreads();

  // layer 1: 2(M) x 4(N) tiles -> exactly one per wave
  {
    int mm = (wv >> 2) << 4, kn = (wv & 3) << 4;
    v8f a = {}, bq = {}, c = {}, d = {};
    wmma_acc_f32(a,  zr + mm * ZP, ZP, w1r + kn, 64, 64);
    wmma_acc_f32(bq, zi + mm * ZP, ZP, w1i + kn, 64, 64);
    wmma_acc_f32(c,  zr + mm * ZP, ZP, w1i + kn, 64, 64);
    wmma_acc_f32(d,  zi + mm * ZP, ZP, w1r + kn, 64, 64);
    int col = kn + l;
    float br = B1r[n * BSZ + col], bi = B1i[n * BSZ + col];
    for (int j = 0; j < 8; ++j) {
      int row = mm + j + h8;
      float vr = a[j] - bq[j] + br;
      float vi = c[j] + d[j] + bi;
      hr[row * ZP + col] = 0.5f * vr * (1.0f + erff(vr * 0.70710678118f));
      hi[row * ZP + col] = 0.5f * vi * (1.0f + erff(vi * 0.70710678118f));
    }
  }
  __syncthreads();

  // layer 2 + softshrink + writeback
  {
    int mm = (wv >> 2) << 4, kn = (wv & 3) << 4;
    v8f a = {}, bq = {}, c = {}, d = {};
    wmma_acc_f32(a,  hr + mm * ZP, ZP, w2r + kn, 64, 64);
    wmma_acc_f32(bq, hi + mm * ZP, ZP, w2i + kn, 64, 64);
    wmma_acc_f32(c,  hr + mm * ZP, ZP, w2i + kn, 64, 64);
    wmma_acc_f32(d,  hi + mm * ZP, ZP, w2r + kn, 64, 64);
    int col = kn + l;
    float br = B2r[n * BSZ + col], bi = B2i[n * BSZ + col];
    for (int j = 0; j < 8; ++j) {
      int row = mm + j + h8;
      float vr = a[j] - bq[j] + br;
      float vi = c[j] + d[j] + bi;
      vr = (vr > LAMBDA) ? vr - LAMBDA : ((vr < -LAMBDA) ? vr + LAMBDA : 0.0f);
      vi = (vi > LAMBDA) ? vi - LAMBDA : ((vi < -LAMBDA) ? vi + LAMBDA : 0.0f);
      int p = p0 + row, k1 = p / M2, k2 = p % M2;
      size_t off = chbase + (size_t)col * (HH * WF) + k1 * WF + k2;
      Zr[off] = vr;
      Zi[off] = vi;
    }
  }
}

// ---------------------------------------------------------------------
// irfft2 (ortho): inverse column DFT (e^{+i}), then hermitian-weighted
// inverse row transform (w_k = 1 for k in {0,64}, else 2), scale 1/128.
// ---------------------------------------------------------------------
__global__ void k_irfft2(const float* __restrict__ Zr, const float* __restrict__ Zi,
                         const float* __restrict__ tab, float* __restrict__ Y)
{
  extern __shared__ float sm[];
  float* zr = sm;                  // 128 x WFP (cols >= 65 zero padded)
  float* zi = zr + HH * WFP;
  float* tr = zi + HH * WFP;
  float* ti = tr + HH * WFP;
  const float* COS = tab;
  const float* SIN = tab + 16384;

  const int img = blockIdx.x;
  const float* zrp = Zr + (size_t)img * (HH * WF);
  const float* zip = Zi + (size_t)img * (HH * WF);
  const int tid = threadIdx.x, wv = tid >> 5;
  const int l = tid & 15, h8 = ((tid >> 4) & 1) * 8;

  for (int i = tid; i < HH * WFL; i += 256) {
    int k1 = i / WFL, k2 = i % WFL;
    bool ok = k2 < WF;
    zr[k1 * WFP + k2] = ok ? zrp[k1 * WF + k2] : 0.0f;
    zi[k1 * WFP + k2] = ok ? zip[k1 * WF + k2] : 0.0f;
  }
  __syncthreads();

  // stage A: t[h][k2] = sum_k1 e^{+2pi i h k1/128} Z[k1][k2], fold w_k/128
  for (int t = wv; t < 40; t += 8) {
    int hm = (t / 5) << 4, kn = (t % 5) << 4;
    v8f a = {}, b = {}, c = {}, d = {};
    wmma_acc_f32(a, COS + hm * 128, 128, zr + kn, WFP, 128);
    wmma_acc_f32(b, SIN + hm * 128, 128, zi + kn, WFP, 128);
    wmma_acc_f32(c, COS + hm * 128, 128, zi + kn, WFP, 128);
    wmma_acc_f32(d, SIN + hm * 128, 128, zr + kn, WFP, 128);
    int col = kn + l;
    float wgt = ((col == 0) || (col == 64)) ? (1.0f / 128.0f) : (2.0f / 128.0f);
    for (int j = 0; j < 8; ++j) {
      int row = hm + j + h8;
      tr[row * WFP + col] = (a[j] - b[j]) * wgt;
      ti[row * WFP + col] = (c[j] + d[j]) * wgt;
    }
  }
  __syncthreads();

  // stage B: y[h][w] = sum_k tr[h][k] cos(2pi wk/N) - ti[h][k] sin(2pi wk/N)
  float* yp = Y + (size_t)img * (HH * WW);
  for (int t = wv; t < 64; t += 8) {
    int hm = (t >> 3) << 4, wn = (t & 7) << 4;
    v8f a = {}, b = {};
    wmma_acc_f32(a, tr + hm * WFP, WFP, COS + wn, 128, WFL);
    wmma_acc_f32(b, ti + hm * WFP, WFP, SIN + wn, 128, WFL);
    for (int j = 0; j < 8; ++j)
      yp[(hm + j + h8) * WW + wn + l] = a[j] - b[j];
  }
}

// ---------------------------------------------------------------------
// 1x1 conv + residual: out[b,o,p] = x[b,o,p] + sum_c F[o,c] * Y[b,c,p]
// 64(O) x 128(pix) block tile, BK=32, double-buffered async LDS staging.
// Buffer pointers are computed at runtime (no LDS-derived pointer arrays,
// which lld rejects as static initializers).
// ---------------------------------------------------------------------
#define FT_SZ (64 * FTP)
#define YT_SZ (32 * 132)

__global__ void k_fuse(const float* __restrict__ x, const float* __restrict__ F,
                       const float* __restrict__ Y, float* __restrict__ out)
{
  extern __shared__ float sm[];

  const int b = blockIdx.z, o0 = blockIdx.y * 64, p0 = blockIdx.x * 128;
  const int tid = threadIdx.x, wv = tid >> 5;
  const int l = tid & 15, h8 = ((tid >> 4) & 1) * 8;
  const int tm = wv >> 1, tn0 = (wv & 1) * 4;

  const float* Yb = Y + (size_t)b * CH * (HH * WW) + p0;
  v8f acc[4] = {};

  auto ftp = [&](int buf) -> float* { return sm + buf * FT_SZ; };
  auto ytp = [&](int buf) -> float* { return sm + 2 * FT_SZ + buf * YT_SZ; };

  auto load_tiles = [&](int buf, int k0) {
    float* ft = ftp(buf);
    float* yt = ytp(buf);
    // F tile: 64 rows x 32 cols = 512 float4 (2 per thread)
    for (int i = tid; i < 512; i += 256) {
      int row = i >> 3, q = (i & 7) * 4;
      ASYNC_CP_F4(ft + row * FTP + q, F + (size_t)(o0 + row) * CH + k0 + q);
    }
    // Y tile: 32 rows x 128 cols = 1024 float4 (4 per thread)
    for (int i = tid; i < 1024; i += 256) {
      int kk = i >> 5, q = (i & 31) * 4;
      ASYNC_CP_F4(yt + kk * 132 + q, Yb + (size_t)(k0 + kk) * (HH * WW) + q);
    }
  };

  load_tiles(0, 0);
  ASYNC_FENCE();
  __syncthreads();

  int cur = 0;
  for (int k0 = 0; k0 < CH; k0 += 32) {
    int nxt = cur ^ 1;
    bool more = (k0 + 32) < CH;
    if (more) load_tiles(nxt, k0 + 32);   // overlap copy with WMMA below
    float* ft = ftp(cur);
    float* yt = ytp(cur);
    for (int t = 0; t < 4; ++t)
      wmma_acc_f32(acc[t], ft + (tm * 16) * FTP, FTP, yt + (tn0 + t) * 16, 132, 32);
    if (more) ASYNC_FENCE();
    __syncthreads();
    cur = nxt;
  }

  for (int t = 0; t < 4; ++t) {
    int col = p0 + (tn0 + t) * 16 + l;
    for (int j = 0; j < 8; ++j) {
      int row = o0 + tm * 16 + j + h8;
      size_t off = ((size_t)b * CH + row) * (HH * WW) + col;
      out[off] = x[off] + acc[t][j];
    }
  }
}

// ---------------------------------------------------------------------
extern "C" void kernel_launch(void* const* d_in, const int* in_sizes, int n_in,
                              void* d_out, int out_size, void* d_ws, size_t ws_size,
                              hipStream_t stream)
{
  (void)in_sizes; (void)n_in; (void)out_size; (void)ws_size;
  const float* x    = (const float*)d_in[0];
  const float* w1r  = (const float*)d_in[1];
  const float* w1i  = (const float*)d_in[2];
  const float* b1r  = (const float*)d_in[3];
  const float* b1i  = (const float*)d_in[4];
  const float* w2r  = (const float*)d_in[5];
  const float* w2i  = (const float*)d_in[6];
  const float* b2r  = (const float*)d_in[7];
  const float* b2i  = (const float*)d_in[8];
  const float* fuse = (const float*)d_in[9];
  float* out = (float*)d_out;

  // workspace: [cos|sin tables 32768] [Zr] [Zi] [Y]
  float* tab = (float*)d_ws;
  float* Zr  = tab + 32768;
  float* Zi  = Zr + (size_t)4 * CH * HH * WF;
  float* Y   = Zi + (size_t)4 * CH * HH * WF;

  k_init_dft<<<dim3(128), dim3(128), 0, stream>>>(tab);

  k_rfft2<<<dim3(4 * CH), dim3(256),
            (HH * XSP + 2 * HH * WFP) * sizeof(float), stream>>>(x, tab, Zr, Zi);

  k_mixer<<<dim3(NPTS / 32, NBK, 4), dim3(256),
            (4 * 4096 + 4 * 32 * ZP) * sizeof(float), stream>>>(
      Zr, Zi, w1r, w1i, b1r, b1i, w2r, w2i, b2r, b2i);

  k_irfft2<<<dim3(4 * CH), dim3(256),
             (4 * HH * WFP) * sizeof(float), stream>>>(Zr, Zi, tab, Y);

  k_fuse<<<dim3(HH * WW / 128, CH / 64, 4), dim3(256),
           (2 * FT_SZ + 2 * YT_SZ) * sizeof(float), stream>>>(
      x, fuse, Y, out);
}